// PhysicsInformedLoss_88828513615950
// MI455X (gfx1250) — compile-verified
//
#include <hip/hip_runtime.h>
#include <hip/hip_bf16.h>

#define LAMBDA_CONT   0.1f
#define LAMBDA_MOM    0.01f
#define INV_REYNOLDS  (1.0f / 1000000.0f)
#define EPS_F         1e-8f

typedef __attribute__((ext_vector_type(2))) float v2f;
typedef __attribute__((ext_vector_type(8))) float v8f;

// ---------------------------------------------------------------------------
// Wave32 sum via V_WMMA_F32_16X16X4_F32.
// A = per-lane value in VGPR0, 0 in VGPR1  (A holds the 32 lane values + zeros)
// B = all-ones 4x16  (layout-independent: every element is 1.0)
// => D[i][j] = rowsum_i(A), and for any column j:  s_j + s_{j^16} = sum(all A)
// One WMMA + 8 VALU adds + 1 shfl_xor gives the exact f32 wave sum in every lane.
// EXEC must be all-1s at the call site (guards only around loads, reconverged here).
// ---------------------------------------------------------------------------
__device__ __forceinline__ float wave_sum32(float v) {
    v2f a; a[0] = v;    a[1] = 0.0f;
    v2f b; b[0] = 1.0f; b[1] = 1.0f;
    v8f c = {0.0f, 0.0f, 0.0f, 0.0f, 0.0f, 0.0f, 0.0f, 0.0f};
    c = __builtin_amdgcn_wmma_f32_16x16x4_f32(
        /*neg_a=*/false, a, /*neg_b=*/false, b,
        /*c_mod=*/(short)0, c, /*reuse_a=*/false, /*reuse_b=*/false);
    float s = c[0] + c[1] + c[2] + c[3] + c[4] + c[5] + c[6] + c[7];
    s += __shfl_xor(s, 16, 32);   // total wave sum in every lane
    return s;
}

// ---------------------------------------------------------------------------
// Workspace layout (floats):
//   S[0]  = sum (pred_vel - target_vel)^2      over all n*3 elements
//   S[1]  = sum (pred_p  - target_p )^2        over all n   elements
//   S[2]  = sum div^2                          over nodes
//   S[3]  = sum residual^2                     over n*3 elements
//   S[16 + node*8 + k] : per-node accumulators
//        k=0: edge count, k=1: sum vel_grad, k=2..4: sum vel_diff/dist_sq,
//        k=5..7: sum p_grad_edge
// ---------------------------------------------------------------------------

__global__ void k_zero(float* __restrict__ p, int n) {
    int i = blockIdx.x * blockDim.x + threadIdx.x;
    if (i < n) p[i] = 0.0f;
}

__global__ void k_data_loss(const float4* __restrict__ pred,
                            const float4* __restrict__ tgt,
                            float* __restrict__ S, int n) {
    int i = blockIdx.x * blockDim.x + threadIdx.x;
    float vs = 0.0f, ps = 0.0f;
    if (i < n) {
        float4 p = pred[i];
        float4 t = tgt[i];
        float dx = p.x - t.x, dy = p.y - t.y, dz = p.z - t.z, dw = p.w - t.w;
        vs = dx * dx + dy * dy + dz * dz;
        ps = dw * dw;
    }
    vs = wave_sum32(vs);
    ps = wave_sum32(ps);
    if ((threadIdx.x & 31) == 0) {
        atomicAdd(S + 0, vs);
        atomicAdd(S + 1, ps);
    }
}

__global__ void k_edges(const int* __restrict__ ei,
                        const float4* __restrict__ pred,
                        const float* __restrict__ pos,
                        float* __restrict__ acc, int nE) {
    int e = blockIdx.x * blockDim.x + threadIdx.x;
    if (e >= nE) return;

    // Prefetch streamed index loads well ahead (global_prefetch_b8);
    // speculative prefetch past the end is dropped safely by hardware.
    __builtin_prefetch(ei + e + 16384, 0, 0);
    __builtin_prefetch(ei + nE + e + 16384, 0, 0);

    int r = ei[e];         // row = edge_index[0][e]
    int c = ei[nE + e];    // col = edge_index[1][e]

    float prx = pos[3 * r + 0], pry = pos[3 * r + 1], prz = pos[3 * r + 2];
    float pcx = pos[3 * c + 0], pcy = pos[3 * c + 1], pcz = pos[3 * c + 2];
    float dx = pcx - prx, dy = pcy - pry, dz = pcz - prz;

    float d2   = dx * dx + dy * dy + dz * dz;
    float dist = sqrtf(d2) + EPS_F;            // norm + EPS
    float inv_dist  = 1.0f / dist;
    float inv_dist2 = inv_dist * inv_dist;     // 1/dist^2 (dist includes EPS)
    float inv_dsq   = 1.0f / (d2 + EPS_F);     // 1/(|d|^2 + EPS)  (separate!)

    float4 qr = pred[r];
    float4 qc = pred[c];
    float vdx = qc.x - qr.x, vdy = qc.y - qr.y, vdz = qc.z - qr.z;
    float pdiff = qc.w - qr.w;

    float vel_grad = (vdx * dx + vdy * dy + vdz * dz) * inv_dist;  // dot(vd, dir)
    float pscale   = pdiff * inv_dist2;        // p_diff/dist * (d/dist) = pdiff*d/dist^2

    float* a = acc + (size_t)r * 8;            // 32B-contiguous per node -> L2 locality
    atomicAdd(a + 0, 1.0f);
    atomicAdd(a + 1, vel_grad);
    atomicAdd(a + 2, vdx * inv_dsq);
    atomicAdd(a + 3, vdy * inv_dsq);
    atomicAdd(a + 4, vdz * inv_dsq);
    atomicAdd(a + 5, pscale * dx);
    atomicAdd(a + 6, pscale * dy);
    atomicAdd(a + 7, pscale * dz);
}

__global__ void k_node_final(const float* __restrict__ acc,
                             float* __restrict__ S, int n) {
    int i = blockIdx.x * blockDim.x + threadIdx.x;
    float cont = 0.0f, mom = 0.0f;
    if (i < n) {
        const float* a = acc + (size_t)i * 8;
        float inv_cnt = 1.0f / fmaxf(a[0], 1.0f);
        float div = a[1] * inv_cnt;
        cont = div * div;
        float r0 = a[2] * inv_cnt * INV_REYNOLDS + a[5] * inv_cnt;
        float r1 = a[3] * inv_cnt * INV_REYNOLDS + a[6] * inv_cnt;
        float r2 = a[4] * inv_cnt * INV_REYNOLDS + a[7] * inv_cnt;
        mom = r0 * r0 + r1 * r1 + r2 * r2;
    }
    cont = wave_sum32(cont);
    mom  = wave_sum32(mom);
    if ((threadIdx.x & 31) == 0) {
        atomicAdd(S + 2, cont);
        atomicAdd(S + 3, mom);
    }
}

__global__ void k_combine(const float* __restrict__ S,
                          float* __restrict__ out, int n) {
    if (blockIdx.x == 0 && threadIdx.x == 0) {
        float fn = (float)n;
        float loss_data = S[0] / (3.0f * fn) + S[1] / fn;
        float loss_cont = S[2] / fn;
        float loss_mom  = S[3] / (3.0f * fn);
        out[0] = loss_data + LAMBDA_CONT * loss_cont + LAMBDA_MOM * loss_mom;
    }
}

extern "C" void kernel_launch(void* const* d_in, const int* in_sizes, int n_in,
                              void* d_out, int out_size, void* d_ws, size_t ws_size,
                              hipStream_t stream) {
    const float4* pred = (const float4*)d_in[0];   // (n,4) f32
    const float4* tgt  = (const float4*)d_in[1];   // (n,4) f32
    const int*    ei   = (const int*)  d_in[2];    // (2,E) i32
    const float*  pos  = (const float*)d_in[3];    // (n,3) f32

    int n  = in_sizes[0] / 4;
    int nE = in_sizes[2] / 2;

    float* S   = (float*)d_ws;   // 16 scalar slots
    float* acc = S + 16;         // 8 floats per node

    int zn = 16 + 8 * n;
    k_zero<<<(zn + 255) / 256, 256, 0, stream>>>(S, zn);
    k_data_loss<<<(n + 255) / 256, 256, 0, stream>>>(pred, tgt, S, n);
    k_edges<<<(nE + 255) / 256, 256, 0, stream>>>(ei, pred, pos, acc, nE);
    k_node_final<<<(n + 255) / 256, 256, 0, stream>>>(acc, S, n);
    k_combine<<<1, 32, 0, stream>>>(S, (float*)d_out, n);
}